// CTCLayer_23862838296849
// MI455X (gfx1250) — compile-verified
//
#include <hip/hip_runtime.h>
#include <stdint.h>

// CTC loss, probability-domain with periodic rescaling.
// B=1024, T=256, C=96 (blank = C-1 = 95), L=32, S = 2L+1 = 65.
// One wave32 per batch element:
//   lane l owns trellis states 2l (blank) and 2l+1 (label l = y_true[b,l]);
//   lane 31 additionally owns state 64 (trailing blank).
// Rows of y_pred[b,t,:] (384 B) are streamed into LDS with
// global_load_async_to_lds_b128 (CDNA5 async path), 8 rows in flight.

#define TT     256
#define CC     96
#define LL     32
#define WAVES  8          // waves per block (one batch element per wave)
#define DEPTH  8          // async rows in flight (power of two)
#define EPSF   1e-7f      // keras epsilon, reference adds it before log

__global__ __launch_bounds__(WAVES * 32)
void ctc_loss_kernel(const int* __restrict__ y_true,
                     const float* __restrict__ y_pred,
                     float* __restrict__ out, int B) {
    __shared__ __align__(16) float rows[WAVES][DEPTH][CC];

    const int lane = threadIdx.x & 31;
    const int w    = threadIdx.x >> 5;
    const int b    = blockIdx.x * WAVES + w;
    if (b >= B) return;

    // --- labels: lane l holds label l of this batch element -----------------
    const int  lab     = y_true[(size_t)b * LL + lane];
    const int  labprev = __shfl_up(lab, 1, 32);
    const bool skip_ok = (lane > 0) && (lab != labprev);   // s-2 -> s allowed

    const float* gbase = y_pred + (size_t)b * TT * CC;

    // --- async row streamer: 24 lanes x 16B = 384B per time step ------------
    auto issue = [&](int row, int slot) {
        if (lane < 24) {
            uint32_t ldsa = (uint32_t)(uintptr_t)&rows[w][slot][lane * 4];
            uint64_t ga   = (uint64_t)(uintptr_t)(gbase + (size_t)row * CC + lane * 4);
            asm volatile("global_load_async_to_lds_b128 %0, %1, off"
                         :: "v"(ldsa), "v"(ga) : "memory");
        }
    };

    #pragma unroll
    for (int r = 0; r < DEPTH; ++r) issue(r, r);

    // row 0 landed once at most DEPTH-1 asyncs remain outstanding
    asm volatile("s_wait_asynccnt 7" ::: "memory");

    // --- t = 0 init: alpha0[0] = p(blank), alpha0[1] = p(label 0) -----------
    float pb = rows[w][0][CC - 1] + EPSF;     // blank prob (LDS broadcast)
    float pl = rows[w][0][lab]    + EPSF;     // own-label prob (LDS gather)
    float a_e  = (lane == 0) ? pb : 0.0f;     // state 2l
    float a_o  = (lane == 0) ? pl : 0.0f;     // state 2l+1
    float a_z  = 0.0f;                        // state 64 (lane 31 only)
    float logZ = 0.0f;

    // buffer slot 0 fully consumed -> safe to overwrite with row DEPTH
    asm volatile("s_wait_dscnt 0" ::: "memory");
    issue(DEPTH, 0);

    // one trellis step over the row in `slot`; caller guarantees row landed
    auto step = [&](int t, int slot, bool refill) {
        pb = rows[w][slot][CC - 1] + EPSF;
        pl = rows[w][slot][lab]    + EPSF;

        // predecessor odd-state alpha from lane-1 (alpha[s-1] / alpha[s-2])
        float po = __shfl_up(a_o, 1, 32);
        if (lane == 0) po = 0.0f;

        float ne = (a_e + po) * pb;                                   // blank
        float no = (a_o + a_e + (skip_ok ? po : 0.0f)) * pl;          // label
        float nz = (a_z + a_o) * pb;                                  // s=64
        a_e = ne;
        a_o = no;
        a_z = (lane == 31) ? nz : 0.0f;

        // refill this slot with row t+DEPTH once our ds reads are done
        if (refill) {
            asm volatile("s_wait_dscnt 0" ::: "memory");
            issue(t + DEPTH, slot);
        }

        // rescale every 4 steps: alpha /= sum, logZ += log(sum)
        if ((t & 3) == 3) {
            float c = a_e + a_o + a_z;
            #pragma unroll
            for (int off = 16; off >= 1; off >>= 1)
                c += __shfl_xor(c, off, 32);
            float inv = 1.0f / c;
            a_e *= inv; a_o *= inv; a_z *= inv;
            logZ += __logf(c);
        }
    };

    // main loop: rows beyond t still outstanding; wait<=DEPTH-1 => row t done
    for (int t = 1; t <= TT - DEPTH; ++t) {
        asm volatile("s_wait_asynccnt 7" ::: "memory");
        step(t, t & (DEPTH - 1), /*refill=*/(t + DEPTH < TT));
    }
    // tail: everything issued; drain and finish the last DEPTH-1 steps
    for (int t = TT - DEPTH + 1; t < TT; ++t) {
        asm volatile("s_wait_asynccnt 0" ::: "memory");
        step(t, t & (DEPTH - 1), /*refill=*/false);
    }

    // loss = -(sum_t log c_t + log(alpha[S-1] + alpha[S-2])); both on lane 31
    if (lane == 31) {
        out[b] = -(logZ + __logf(a_o + a_z));
    }
}

extern "C" void kernel_launch(void* const* d_in, const int* in_sizes, int n_in,
                              void* d_out, int out_size, void* d_ws, size_t ws_size,
                              hipStream_t stream) {
    const int*   y_true = (const int*)d_in[0];     // [B, L] int32
    const float* y_pred = (const float*)d_in[1];   // [B, T, C] f32 probabilities
    float*       out    = (float*)d_out;           // [B] f32 loss
    (void)d_ws; (void)ws_size; (void)n_in; (void)out_size;

    const int B = in_sizes[0] / LL;
    dim3 grid((B + WAVES - 1) / WAVES);
    dim3 block(WAVES * 32);
    hipLaunchKernelGGL(ctc_loss_kernel, grid, block, 0, stream,
                       y_true, y_pred, out, B);
}